// QuadrupletLoss_43001212567640
// MI455X (gfx1250) — compile-verified
//
#include <hip/hip_runtime.h>
#include <hip/hip_bf16.h>

typedef float v2f __attribute__((ext_vector_type(2)));
typedef float v8f __attribute__((ext_vector_type(8)));

#define BN     8192
#define DDIM   512
#define NTILES (BN / 16)   // 512 column tiles
#define NBLK   (BN / 64)   // 128 row blocks (4 A tiles each)

static constexpr float kM1  = 0.25f;
static constexpr float kM2  = 0.15f;
static constexpr float kBIG = 1e30f;
static constexpr float kHAS = 1e29f;   // "min was updated at least once"

// ---------------------------------------------------------------------------
// Kernel 1: per-row squared norm + packed class key. One wave per row.
// key = (coarse << 7) | fine.  Since coarse = f2c[fine] is a function of fine:
//   same_fine(i,j)   <=>  key_i == key_j
//   same_coarse(i,j) <=>  (key_i ^ key_j) < 128
// ---------------------------------------------------------------------------
__global__ __launch_bounds__(256) void rowstats_kernel(
    const float* __restrict__ X, const int* __restrict__ tf,
    const int* __restrict__ f2c, float* __restrict__ sq,
    unsigned* __restrict__ key) {
  const int wave = threadIdx.x >> 5;
  const int lane = threadIdx.x & 31;
  const int row  = blockIdx.x * 8 + wave;
  const float4* xr = (const float4*)(X + (size_t)row * DDIM);
  float s = 0.0f;
#pragma unroll
  for (int q = 0; q < 4; ++q) {
    float4 v = xr[lane + q * 32];
    s += v.x * v.x + v.y * v.y + v.z * v.z + v.w * v.w;
  }
#pragma unroll
  for (int m = 16; m >= 1; m >>= 1) s += __shfl_xor(s, m, 32);
  if (lane == 0) {
    const int f = tf[row];
    sq[row]  = s;
    key[row] = ((unsigned)f2c[f] << 7) | (unsigned)f;
  }
}

// ---------------------------------------------------------------------------
// Kernel 2: first same-fine index (p_plus) and its distance. One block per anchor.
// ---------------------------------------------------------------------------
__global__ __launch_bounds__(256) void firstpos_kernel(
    const float* __restrict__ X, const int* __restrict__ tf,
    const float* __restrict__ sq, float* __restrict__ dist_pos,
    int* __restrict__ has_pos) {
  __shared__ int   s_min[256];
  __shared__ float s_dot[256];
  const int i   = blockIdx.x;
  const int t   = threadIdx.x;
  const int tfi = tf[i];

  int best = 0x7fffffff;
  for (int j = t; j < BN; j += 256) {          // ascending => first hit is this
    if (tf[j] == tfi && j != i) { best = j; break; }  // thread's minimum
  }
  s_min[t] = best;
  __syncthreads();
  for (int off = 128; off > 0; off >>= 1) {
    if (t < off) s_min[t] = min(s_min[t], s_min[t + off]);
    __syncthreads();
  }
  const int p = s_min[0];

  float dot = 0.0f;
  if (p != 0x7fffffff) {
    const float* xi = X + (size_t)i * DDIM;
    const float* xp = X + (size_t)p * DDIM;
    dot = xi[t] * xp[t] + xi[t + 256] * xp[t + 256];
  }
  s_dot[t] = dot;
  __syncthreads();
  for (int off = 128; off > 0; off >>= 1) {
    if (t < off) s_dot[t] += s_dot[t + off];
    __syncthreads();
  }
  if (t == 0) {
    if (p == 0x7fffffff) {
      has_pos[i]  = 0;
      dist_pos[i] = 0.0f;
    } else {
      has_pos[i]  = 1;
      dist_pos[i] = fmaxf(sq[i] + sq[p] - 2.0f * s_dot[0], 0.0f);
    }
  }
}

// ---------------------------------------------------------------------------
// Kernel 3: fused X·X^T (fp32 WMMA) + masked row-min reduction + per-row loss.
// One workgroup (8 waves) owns FOUR 16-row A tiles (64 rows, 128 KB LDS);
// waves split the 512 column tiles. Each B fragment (global_load_b64) feeds
// FOUR WMMAs -> 0.25 global-b64 per WMMA; A comes from LDS as merged
// ds_load_2addr_b64. Four independent accumulator chains (one per tile).
// ---------------------------------------------------------------------------
__global__ __launch_bounds__(256) void quad_main_kernel(
    const float* __restrict__ X, const unsigned* __restrict__ key,
    const float* __restrict__ sq, const float* __restrict__ dist_pos,
    const int* __restrict__ has_pos, float* __restrict__ partial) {
  __shared__ __align__(16) float s_a[64 * DDIM];   // 128 KB: four A tiles
  __shared__ float    s_sq[64];
  __shared__ unsigned s_key[64];
  __shared__ float    s_pm[8][64];
  __shared__ float    s_ng[8][64];
  __shared__ float    s_loss[64];
  __shared__ float    s_cnt[64];

  const int tid     = threadIdx.x;
  const int wave    = tid >> 5;
  const int lane    = tid & 31;
  const int lane15  = lane & 15;
  const int half    = lane >> 4;           // selects K={0,1} vs K={2,3} stream
  const int rowbase = blockIdx.x * 64;

  // Stage the 64x512 fp32 A tiles into LDS (coalesced float4).
  {
    const float4* src = (const float4*)(X + (size_t)rowbase * DDIM);
    float4*       dst = (float4*)s_a;
#pragma unroll
    for (int q = 0; q < 32; ++q) dst[tid + q * 256] = src[tid + q * 256];
  }
  if (tid < 64) {
    const int r = rowbase + tid;
    s_sq[tid]  = sq[r];
    s_key[tid] = key[r];
  }
  __syncthreads();

  // Per-lane row attributes: tile t, acc register r -> row (t*16 + r + 8*half).
  float    sq_i[4][8];
  unsigned key_i[4][8];
#pragma unroll
  for (int t = 0; t < 4; ++t)
#pragma unroll
    for (int r = 0; r < 8; ++r) {
      const int rr = t * 16 + r + 8 * half;
      sq_i[t][r]  = s_sq[rr];
      key_i[t][r] = s_key[rr];
    }

  float pmmin[4][8], ngmin[4][8];
#pragma unroll
  for (int t = 0; t < 4; ++t)
#pragma unroll
    for (int r = 0; r < 8; ++r) { pmmin[t][r] = kBIG; ngmin[t][r] = kBIG; }

  // A-fragment streams (fp32 16x16x4 layout: lane<16 -> K=0,1; lane>=16 -> K=2,3).
  const float* aptr[4];
#pragma unroll
  for (int t = 0; t < 4; ++t)
    aptr[t] = s_a + (size_t)(t * 16 + lane15) * DDIM + 2 * half;

  for (int jt = wave; jt < NTILES; jt += 8) {
    const int      col   = jt * 16 + lane15;
    const float*   bptr  = X + (size_t)col * DDIM + 2 * half;  // shared B stream
    const float    sq_j  = sq[col];
    const unsigned key_j = key[col];

    // Prefetch the next column tile's B row for this lane (global_prefetch_b8).
    if (jt + 8 < NTILES) {
      __builtin_prefetch(bptr + 8 * 16 * DDIM, 0, 3);
      __builtin_prefetch(bptr + 8 * 16 * DDIM + 128, 0, 3);
    }

    v8f acc[4] = {{}, {}, {}, {}};   // one chain per tile (4-issue spacing)
#pragma unroll 4
    for (int k = 0; k < DDIM; k += 4) {
      v2f b = *(const v2f*)(bptr + k);
#pragma unroll
      for (int t = 0; t < 4; ++t) {
        v2f a = *(const v2f*)(aptr[t] + k);
        acc[t] = __builtin_amdgcn_wmma_f32_16x16x4_f32(false, a, false, b,
                                                       (short)0, acc[t],
                                                       false, false);
      }
    }

#pragma unroll
    for (int t = 0; t < 4; ++t)
#pragma unroll
      for (int r = 0; r < 8; ++r) {
        const float    d = fmaxf(fmaf(-2.0f, acc[t][r], sq_i[t][r] + sq_j), 0.0f);
        const unsigned x = key_i[t][r] ^ key_j;
        if (x != 0u && x < 128u) pmmin[t][r] = fminf(pmmin[t][r], d);  // pm_mask
        if (x >= 128u)           ngmin[t][r] = fminf(ngmin[t][r], d);  // neg_mask
      }
  }

  // Reduce the running mins across each 16-lane group (row lives in a half-wave).
#pragma unroll
  for (int t = 0; t < 4; ++t)
#pragma unroll
    for (int r = 0; r < 8; ++r) {
      float pm = pmmin[t][r], ng = ngmin[t][r];
#pragma unroll
      for (int m = 8; m >= 1; m >>= 1) {
        pm = fminf(pm, __shfl_xor(pm, m, 32));
        ng = fminf(ng, __shfl_xor(ng, m, 32));
      }
      if (lane15 == 0) {
        s_pm[wave][t * 16 + r + 8 * half] = pm;
        s_ng[wave][t * 16 + r + 8 * half] = ng;
      }
    }
  __syncthreads();

  // Cross-wave min + per-row loss; deterministic per-block partial sum.
  if (tid < 64) {
    float pm = kBIG, ng = kBIG;
#pragma unroll
    for (int w = 0; w < 8; ++w) {
      pm = fminf(pm, s_pm[w][tid]);
      ng = fminf(ng, s_ng[w][tid]);
    }
    const int   i  = rowbase + tid;
    const float dp = dist_pos[i];
    const bool  hp  = has_pos[i] != 0;
    const bool  hpm = pm < kHAS;
    const bool  hng = ng < kHAS;
    const float in_loss  = fmaxf(dp - pm + (kM1 - kM2), 0.0f);
    const float out_loss = fmaxf(pm - ng + kM2, 0.0f);
    const bool  valid = hp && hpm && hng && (in_loss > 0.0f) && (out_loss > 0.0f);
    s_loss[tid] = valid ? (in_loss + out_loss) : 0.0f;
    s_cnt[tid]  = valid ? 1.0f : 0.0f;
  }
  __syncthreads();
  if (tid == 0) {
    float ls = 0.0f, cn = 0.0f;
#pragma unroll
    for (int r = 0; r < 64; ++r) { ls += s_loss[r]; cn += s_cnt[r]; }
    partial[2 * blockIdx.x]     = ls;
    partial[2 * blockIdx.x + 1] = cn;
  }
}

// ---------------------------------------------------------------------------
// Kernel 4: deterministic final reduction.
// ---------------------------------------------------------------------------
__global__ void finalize_kernel(const float* __restrict__ partial,
                                float* __restrict__ out) {
  float s = 0.0f, n = 0.0f;
  for (int b = 0; b < NBLK; ++b) {
    s += partial[2 * b];
    n += partial[2 * b + 1];
  }
  out[0] = (n > 0.0f) ? s / (2.0f * n) : 0.0f;
}

// ---------------------------------------------------------------------------
extern "C" void kernel_launch(void* const* d_in, const int* in_sizes, int n_in,
                              void* d_out, int out_size, void* d_ws, size_t ws_size,
                              hipStream_t stream) {
  const float* X   = (const float*)d_in[0];   // [8192, 512] fp32
  const int*   tf  = (const int*)d_in[1];     // [8192]
  const int*   f2c = (const int*)d_in[2];     // [100]
  float*       out = (float*)d_out;           // [1]

  float*    ws       = (float*)d_ws;
  float*    sq       = ws;                                 // 8192 f
  float*    dist_pos = ws + BN;                            // 8192 f
  float*    partial  = ws + 2 * BN;                        // 2*128 f
  unsigned* key      = (unsigned*)(ws + 2 * BN + 2 * NBLK);// 8192 u32
  int*      has_pos  = (int*)(key + BN);                   // 8192 i

  rowstats_kernel<<<BN / 8, 256, 0, stream>>>(X, tf, f2c, sq, key);
  firstpos_kernel<<<BN, 256, 0, stream>>>(X, tf, sq, dist_pos, has_pos);
  quad_main_kernel<<<NBLK, 256, 0, stream>>>(X, key, sq, dist_pos,
                                             has_pos, partial);
  finalize_kernel<<<1, 1, 0, stream>>>(partial, out);
}